// GraphAttentionLayer_43559558316887
// MI455X (gfx1250) — compile-verified
//
#include <hip/hip_runtime.h>
#include <hip/hip_bf16.h>

#define GAT_B    2
#define GAT_N    50000
#define GAT_E    800000
#define GAT_D    128           // DIN == DOUT == 128
#define GAT_M    (GAT_B * GAT_N)
#define GAT_ALPHA 0.2f

typedef float v2f __attribute__((ext_vector_type(2)));
typedef float v8f __attribute__((ext_vector_type(8)));

// ---------- order-preserving float<->uint encoding for atomic max ----------
__device__ __forceinline__ unsigned encf(float f) {
    unsigned u = __float_as_uint(f);
    return (u & 0x80000000u) ? ~u : (u | 0x80000000u);
}
__device__ __forceinline__ float decf(unsigned u) {
    return __uint_as_float((u & 0x80000000u) ? (u ^ 0x80000000u) : ~u);
}
// enc(-inf) = ~0xFF800000 = 0x007FFFFF
#define ENC_NEG_INF 0x007FFFFFu

// ---------------------------------------------------------------------------
// Kernel 0: initialize output accumulator, segment-max, denominators
// ---------------------------------------------------------------------------
__global__ void gat_init(float* __restrict__ out, size_t nout,
                         unsigned* __restrict__ menc, float* __restrict__ denom,
                         int nnodes) {
    size_t i = (size_t)blockIdx.x * blockDim.x + threadIdx.x;
    if (i < nout) out[i] = 0.0f;
    if (i < (size_t)nnodes) {
        menc[i]  = ENC_NEG_INF;
        denom[i] = 0.0f;
    }
}

// ---------------------------------------------------------------------------
// Kernel 1: Wh = h @ W_fc^T via V_WMMA_F32_16X16X4_F32
//   Each wave owns a 16(M) x 128(N) slab. W_fc staged in LDS (64 KB).
//   A frag (16x4 f32):  lane<16 -> row m0+lane, K=k..k+1 ; lane>=16 -> K=k+2..k+3
//   B frag (4x16 f32):  B[kk][n] = W_fc[n][kk], mirrored half-wave K split
//   C/D (16x16 f32):    VGPR r -> row m0 + r + 8*(lane>=16), col n0 + (lane&15)
// ---------------------------------------------------------------------------
__global__ __launch_bounds__(256) void gat_wh_gemm(
        const float* __restrict__ h, const float* __restrict__ Wfc,
        float* __restrict__ Wh, int Mrows) {
    __shared__ __align__(16) float sW[GAT_D * GAT_D];   // 64 KB: W_fc row-major [o][k]
    for (int i = threadIdx.x; i < GAT_D * GAT_D; i += blockDim.x) sW[i] = Wfc[i];
    __syncthreads();

    const int lane = threadIdx.x & 31;
    const int wave = threadIdx.x >> 5;
    const int m0   = (blockIdx.x * 8 + wave) * 16;
    if (m0 >= Mrows) return;                 // whole-wave exit, EXEC stays all-1s

    const int half = lane >> 4;              // 0: K,K+1   1: K+2,K+3
    const int l16  = lane & 15;
    const float* hrow = h + (size_t)(m0 + l16) * GAT_D;

    v8f acc[8] = {};                         // 8 N-tiles of 16

    for (int k = 0; k < GAT_D; k += 4) {
        v2f a = *(const v2f*)(hrow + k + half * 2);
#pragma unroll
        for (int t = 0; t < 8; ++t) {
            const float* wrow = sW + (t * 16 + l16) * GAT_D;
            v2f b = *(const v2f*)(wrow + k + half * 2);   // ds_load_b64
            acc[t] = __builtin_amdgcn_wmma_f32_16x16x4_f32(
                         false, a, false, b, (short)0, acc[t], false, false);
        }
    }

#pragma unroll
    for (int t = 0; t < 8; ++t) {
        const int col = t * 16 + l16;
#pragma unroll
        for (int r = 0; r < 8; ++r) {
            const int row = m0 + r + half * 8;
            Wh[(size_t)row * GAT_D + col] = acc[t][r];
        }
    }
}

// ---------------------------------------------------------------------------
// Kernel 2: s_src[row] = Wh[row]·a_src, s_dst[row] = Wh[row]·a_dst
//   one wave32 per row, float4 per lane, shuffle reduction
// ---------------------------------------------------------------------------
__global__ __launch_bounds__(256) void gat_attn_scores(
        const float* __restrict__ Wh, const float* __restrict__ Wattn,
        float* __restrict__ s_src, float* __restrict__ s_dst, int Mrows) {
    const int lane = threadIdx.x & 31;
    const int row  = (int)(((size_t)blockIdx.x * blockDim.x + threadIdx.x) >> 5);
    if (row >= Mrows) return;

    const float4 w  = *(const float4*)(Wh + (size_t)row * GAT_D + lane * 4);
    const float4 as = *(const float4*)(Wattn + lane * 4);
    const float4 ad = *(const float4*)(Wattn + GAT_D + lane * 4);
    float ps = w.x * as.x + w.y * as.y + w.z * as.z + w.w * as.w;
    float pd = w.x * ad.x + w.y * ad.y + w.z * ad.z + w.w * ad.w;
#pragma unroll
    for (int off = 16; off; off >>= 1) {
        ps += __shfl_xor(ps, off, 32);
        pd += __shfl_xor(pd, off, 32);
    }
    if (lane == 0) { s_src[row] = ps; s_dst[row] = pd; }
}

// ---------------------------------------------------------------------------
// Kernel 3: per-edge leaky-relu score + segment max over dst (both batches)
// ---------------------------------------------------------------------------
__global__ void gat_edge_scores(const int* __restrict__ ei,
                                const float* __restrict__ s_src,
                                const float* __restrict__ s_dst,
                                float* __restrict__ escore,
                                unsigned* __restrict__ menc) {
    const int e = blockIdx.x * blockDim.x + threadIdx.x;
    if (e >= GAT_E) return;
    const int s = ei[e], d = ei[GAT_E + e];
#pragma unroll
    for (int b = 0; b < GAT_B; ++b) {
        float v = s_src[b * GAT_N + s] + s_dst[b * GAT_N + d];
        v = (v > 0.0f) ? v : GAT_ALPHA * v;
        escore[(size_t)b * GAT_E + e] = v;
        atomicMax(&menc[b * GAT_N + d], encf(v));
    }
}

// ---------------------------------------------------------------------------
// Kernel 4: ex = exp(score - max[dst]); denom[dst] += ex
// ---------------------------------------------------------------------------
__global__ void gat_edge_exp(const int* __restrict__ ei,
                             const unsigned* __restrict__ menc,
                             float* __restrict__ escore,
                             float* __restrict__ denom) {
    const int e = blockIdx.x * blockDim.x + threadIdx.x;
    if (e >= GAT_E) return;
    const int d = ei[GAT_E + e];
#pragma unroll
    for (int b = 0; b < GAT_B; ++b) {
        const float m  = decf(menc[b * GAT_N + d]);
        const float ex = __expf(escore[(size_t)b * GAT_E + e] - m);
        escore[(size_t)b * GAT_E + e] = ex;
        atomicAdd(&denom[b * GAT_N + d], ex);
    }
}

// ---------------------------------------------------------------------------
// Kernel 5: out[b,dst,:] += (ex/denom[dst]) * Wh[b,src,:]   (wave per edge)
// ---------------------------------------------------------------------------
__global__ __launch_bounds__(256) void gat_scatter_msg(
        const int* __restrict__ ei, const float* __restrict__ escore,
        const float* __restrict__ denom, const float* __restrict__ Wh,
        float* __restrict__ out) {
    const int lane = threadIdx.x & 31;
    const long long e = (long long)(((size_t)blockIdx.x * blockDim.x + threadIdx.x) >> 5);
    if (e >= GAT_E) return;
    const int s = ei[e], d = ei[GAT_E + e];
#pragma unroll
    for (int b = 0; b < GAT_B; ++b) {
        const float attn = escore[(size_t)b * GAT_E + e] / denom[b * GAT_N + d];
        const float4 w = *(const float4*)(Wh + ((size_t)b * GAT_N + s) * GAT_D + lane * 4);
        float* o = out + ((size_t)b * GAT_N + d) * GAT_D + lane * 4;
        atomicAdd(o + 0, attn * w.x);
        atomicAdd(o + 1, attn * w.y);
        atomicAdd(o + 2, attn * w.z);
        atomicAdd(o + 3, attn * w.w);
    }
}

// ---------------------------------------------------------------------------
// Kernel 6: ELU in place (float4)
// ---------------------------------------------------------------------------
__global__ void gat_elu(float* __restrict__ out, size_t n4) {
    const size_t i = (size_t)blockIdx.x * blockDim.x + threadIdx.x;
    if (i >= n4) return;
    float4 v = ((float4*)out)[i];
    v.x = v.x > 0.0f ? v.x : (__expf(v.x) - 1.0f);
    v.y = v.y > 0.0f ? v.y : (__expf(v.y) - 1.0f);
    v.z = v.z > 0.0f ? v.z : (__expf(v.z) - 1.0f);
    v.w = v.w > 0.0f ? v.w : (__expf(v.w) - 1.0f);
    ((float4*)out)[i] = v;
}

// ---------------------------------------------------------------------------
extern "C" void kernel_launch(void* const* d_in, const int* in_sizes, int n_in,
                              void* d_out, int out_size, void* d_ws, size_t ws_size,
                              hipStream_t stream) {
    const float* h     = (const float*)d_in[0];   // (B,N,128) f32
    const int*   ei    = (const int*)  d_in[1];   // (2,E) i32
    const float* Wfc   = (const float*)d_in[2];   // (128,128) f32
    const float* Wattn = (const float*)d_in[3];   // (1,256) f32
    float* out = (float*)d_out;                   // (B,N,128) f32

    // workspace layout (floats)
    float*    Wh     = (float*)d_ws;                        // 12,800,000
    float*    s_src  = Wh + (size_t)GAT_M * GAT_D;          //    100,000
    float*    s_dst  = s_src + GAT_M;                       //    100,000
    unsigned* menc   = (unsigned*)(s_dst + GAT_M);          //    100,000
    float*    denom  = (float*)(menc + GAT_M);              //    100,000
    float*    escore = denom + GAT_M;                       //  1,600,000

    const size_t nout = (size_t)GAT_M * GAT_D;              // 12.8M floats

    // 0: init accumulators
    gat_init<<<(int)((nout + 255) / 256), 256, 0, stream>>>(out, nout, menc, denom, GAT_M);

    // 1: Wh GEMM (WMMA). 6250 M-tiles of 16, 8 waves/block.
    gat_wh_gemm<<<(GAT_M / 16 + 7) / 8, 256, 0, stream>>>(h, Wfc, Wh, GAT_M);

    // 2: per-row attention dot products (one wave per row)
    gat_attn_scores<<<(GAT_M * 32 + 255) / 256, 256, 0, stream>>>(Wh, Wattn, s_src, s_dst, GAT_M);

    // 3: edge scores + segment max
    gat_edge_scores<<<(GAT_E + 255) / 256, 256, 0, stream>>>(ei, s_src, s_dst, escore, menc);

    // 4: exp + denominator
    gat_edge_exp<<<(GAT_E + 255) / 256, 256, 0, stream>>>(ei, menc, escore, denom);

    // 5: message scatter (one wave per edge)
    gat_scatter_msg<<<(int)(((size_t)GAT_E * 32 + 255) / 256), 256, 0, stream>>>(
        ei, escore, denom, Wh, out);

    // 6: ELU in place
    gat_elu<<<(int)((nout / 4 + 255) / 256), 256, 0, stream>>>(out, nout / 4);
}